// MPC_17446157157007
// MI455X (gfx1250) — compile-verified
//
#include <hip/hip_runtime.h>
#include <stdint.h>

// Batched LQR (Riccati backward + forward rollout) for MI455X / gfx1250.
// One wave32 per batch element; FP32 WMMA (v_wmma_f32_16x16x4_f32) for all
// matrix products; async Global->LDS staging of F_t (ASYNCcnt pipe);
// in-LDS Gauss-Jordan for the 16x16 SPD solve; K/k staged through d_ws
// (needs 256*50*528*4 = ~27 MB).

#define Tn  50
#define Bn  256
#define NST 32
#define NC  16
#define NS  48
#define KSTRIDE (NC*NST + NC)   // 528 floats per (b,t): K row-major + kk

typedef float v2f __attribute__((ext_vector_type(2)));
typedef float v8f __attribute__((ext_vector_type(8)));

__device__ __forceinline__ v8f wmma4(v2f a, v2f b, v8f c) {
  // D = A(16x4) * B(4x16) + C, fp32 throughout.
  return __builtin_amdgcn_wmma_f32_16x16x4_f32(false, a, false, b,
                                               (short)0, c, false, false);
}

// Async copy: 32 lanes x 16B = 512B per issue, global -> LDS, ASYNCcnt pipe.
__device__ __forceinline__ void async_b128_to_lds(uint32_t lds_addr,
                                                  uint32_t goff,
                                                  const void* gbase) {
  asm volatile("global_load_async_to_lds_b128 %0, %1, %2 offset:0"
               :: "v"(lds_addr), "v"(goff), "s"(gbase) : "memory");
}
__device__ __forceinline__ void wait_async0() {
  asm volatile("s_wait_asynccnt 0x0" ::: "memory");
}

// A operand: logical A[16x4] tile at (r0, c0) of row-major src (leading dim ld).
// Lane layout (ISA 7.12.2): lanes 0-15 -> K = c0+{0,1}; lanes 16-31 -> K = c0+{2,3}.
__device__ __forceinline__ v2f ldA(const float* src, int ld, int r0, int c0,
                                   int lm, int lh) {
  const int k0 = c0 + 2*lh;
  v2f a;
  a.x = src[(r0 + lm)*ld + k0 + 0];
  a.y = src[(r0 + lm)*ld + k0 + 1];
  return a;
}

// A operand where the logical matrix is srcᵀ (A[m][k] = src[k][m]).
__device__ __forceinline__ v2f ldAT(const float* src, int ld, int r0, int c0,
                                    int lm, int lh) {
  const int k0 = c0 + 2*lh;
  v2f a;
  a.x = src[(k0 + 0)*ld + r0 + lm];
  a.y = src[(k0 + 1)*ld + r0 + lm];
  return a;
}

// B operand: logical B[4x16] tile, rows (K) at r0, cols at c0.
__device__ __forceinline__ v2f ldB(const float* src, int ld, int r0, int c0,
                                   int lm, int lh) {
  const int k0 = r0 + 2*lh;
  v2f b;
  b.x = src[(k0 + 0)*ld + c0 + lm];
  b.y = src[(k0 + 1)*ld + c0 + lm];
  return b;
}

// C/D 16x16 tile <-> row-major matrix. VGPR r: lanes 0-15 M=r, lanes 16-31 M=r+8.
__device__ __forceinline__ v8f load_tile(const float* src, int ld, int r0, int c0,
                                         int lm, int lh) {
  v8f c;
#pragma unroll
  for (int r = 0; r < 8; ++r)
    c[r] = src[(size_t)(r0 + r + 8*lh)*ld + c0 + lm];
  return c;
}

__device__ __forceinline__ void store_tile(float* dst, int ld, int r0, int c0,
                                           int lm, int lh, v8f c) {
#pragma unroll
  for (int r = 0; r < 8; ++r)
    dst[(size_t)(r0 + r + 8*lh)*ld + c0 + lm] = c[r];
}

__global__ __launch_bounds__(32)
void lqr_riccati_kernel(const float* __restrict__ Cg,   // (T,B,48,48)
                        const float* __restrict__ cg,   // (T,B,48)
                        const float* __restrict__ Fg,   // (T-1,B,32,48)
                        const float* __restrict__ fg,   // (T-1,B,32)
                        const float* __restrict__ x0g,  // (B,32)
                        float* __restrict__ tau,        // (T,B,48)
                        float* __restrict__ ws) {       // K/k scratch
  const int b    = blockIdx.x;
  const int lane = threadIdx.x;        // wave32: 0..31
  const int lm   = lane & 15;
  const int lh   = lane >> 4;

  __shared__ float Vs[NST*NST];        // value Hessian V_{t+1}
  __shared__ float Vt[NST*NST];        // unsymmetrized V'
  __shared__ float Fs[NST*NS];         // F_t (async-staged)
  __shared__ float Ws[NST*NS];         // W = V * F
  __shared__ float Qs[NS*NS];          // Q
  __shared__ float As[NC*NC];          // Quu (symmetrized), Gauss-Jordan LHS
  __shared__ float Rs[NC*34];          // [Qux | qu] -> solution -> -solution = [K | k]
  __shared__ float zv[NST];            // V f + v
  __shared__ float vv[NST];            // v_{t+1}
  __shared__ float fv[NST];            // f_t
  __shared__ float qv[NS];             // q
  __shared__ float xv[NST];            // forward state x
  __shared__ float xu[NS];             // [x; u]

  // Generic shared pointers carry the LDS byte offset in their low 32 bits
  // (flat LDS aperture).
  const uint32_t Fs_lds = (uint32_t)(uintptr_t)(void*)&Fs[0];

  // ---- init carry V=0, v=0 ----
  for (int i = lane; i < NST*NST; i += 32) Vs[i] = 0.f;
  if (lane < NST) vv[lane] = 0.f;
  __syncthreads();

  // ================= backward Riccati recursion =================
  for (int t = Tn - 1; t >= 0; --t) {
    const float* Ct = Cg + ((size_t)t*Bn + b)*NS*NS;
    const float* ct = cg + ((size_t)t*Bn + b)*NS;
    const bool hasF = (t < Tn - 1);

    if (hasF) {
      const float* Ft = Fg + ((size_t)t*Bn + b)*NST*NS;
      const float* ft = fg + ((size_t)t*Bn + b)*NST;
      // Async-stage F_t (6 KB): 12 x (32 lanes x 16B) issues on ASYNCcnt.
#pragma unroll
      for (int off = 0; off < NST*NS*4; off += 32*16) {
        const uint32_t o = (uint32_t)off + (uint32_t)lane*16u;
        async_b128_to_lds(Fs_lds + o, o, (const void*)Ft);
      }
      if (lane < NST) fv[lane] = ft[lane];
    }

    // Prefetch next (t-1) step's C and F while we crunch this one
    // (serial chain per wave -> hide HBM latency behind compute).
    if (t > 0) {
      const char* cp = (const char*)(Cg + ((size_t)(t-1)*Bn + b)*NS*NS);
      for (int off = lane*128; off < NS*NS*4; off += 32*128)
        __builtin_prefetch(cp + off, 0, 1);
      if (t - 1 < Tn - 1) {
        const char* fp = (const char*)(Fg + ((size_t)(t-1)*Bn + b)*NST*NS);
        for (int off = lane*128; off < NST*NS*4; off += 32*128)
          __builtin_prefetch(fp + off, 0, 1);
      }
    }

    if (hasF) wait_async0();   // F_t resident in LDS
    __syncthreads();

    if (hasF) {
      // ---- W = V (32x32) @ F (32x48) ----
      for (int mi = 0; mi < 2; ++mi)
        for (int nj = 0; nj < 3; ++nj) {
          v8f acc = {0.f,0.f,0.f,0.f,0.f,0.f,0.f,0.f};
#pragma unroll
          for (int kc = 0; kc < 8; ++kc) {
            v2f a  = ldA(Vs, NST, 16*mi, 4*kc, lm, lh);
            v2f bb = ldB(Fs, NS,  4*kc, 16*nj, lm, lh);
            acc = wmma4(a, bb, acc);
          }
          store_tile(Ws, NS, 16*mi, 16*nj, lm, lh, acc);
        }
      // ---- z = V f + v ----
      if (lane < NST) {
        float s = vv[lane];
        for (int k2 = 0; k2 < NST; ++k2) s += Vs[lane*NST + k2] * fv[k2];
        zv[lane] = s;
      }
    }
    __syncthreads();

    // ---- q = c + Fᵀ z ----
    for (int base = 0; base < NS; base += 32) {
      const int i = base + lane;
      if (i < NS) {
        float s = ct[i];
        if (hasF)
          for (int j = 0; j < NST; ++j) s += Fs[j*NS + i] * zv[j];
        qv[i] = s;
      }
    }

    // ---- Q = C + Fᵀ W  (48x48, 9 tiles, K=32) ----
    for (int ti = 0; ti < 3; ++ti)
      for (int tj = 0; tj < 3; ++tj) {
        v8f acc = load_tile(Ct, NS, 16*ti, 16*tj, lm, lh);
        if (hasF) {
#pragma unroll
          for (int kc = 0; kc < 8; ++kc) {
            v2f a  = ldAT(Fs, NS, 16*ti, 4*kc, lm, lh);   // A = Fᵀ
            v2f bb = ldB (Ws, NS, 4*kc, 16*tj, lm, lh);
            acc = wmma4(a, bb, acc);
          }
        }
        store_tile(Qs, NS, 16*ti, 16*tj, lm, lh, acc);
      }
    __syncthreads();

    // ---- Quu symmetrize; build [Qux | qu] ----
    for (int idx = lane; idx < NC*NC; idx += 32) {
      const int r = idx >> 4, cc = idx & 15;
      As[idx] = 0.5f*(Qs[(NST+r)*NS + NST+cc] + Qs[(NST+cc)*NS + NST+r]);
    }
    for (int idx = lane; idx < NC*NST; idx += 32) {
      const int r = idx >> 5, cc = idx & 31;
      Rs[r*34 + cc] = Qs[(NST+r)*NS + cc];
    }
    if (lane < NC) Rs[lane*34 + 32] = qv[NST + lane];
    __syncthreads();

    // ---- Gauss-Jordan: Quu X = [Qux | qu] (SPD, no pivoting) ----
    for (int k = 0; k < NC; ++k) {
      const float inv = 1.0f / As[k*NC + k];
      if (lane < NC) As[k*NC + lane] *= inv;
      for (int j = lane; j < 33; j += 32) Rs[k*34 + j] *= inv;
      __syncthreads();
      for (int i = 0; i < NC; ++i) {
        if (i == k) continue;
        const float fct = As[i*NC + k];
        __syncthreads();
        if (lane < NC) As[i*NC + lane] -= fct * As[k*NC + lane];
        for (int j = lane; j < 33; j += 32) Rs[i*34 + j] -= fct * Rs[k*34 + j];
        __syncthreads();
      }
    }

    // ---- negate: Rs = [K | k] = -X; spill to ws for the forward pass ----
    for (int idx = lane; idx < NC*34; idx += 32) Rs[idx] = -Rs[idx];
    __syncthreads();
    float* kws = ws + ((size_t)b*Tn + t)*KSTRIDE;
    for (int idx = lane; idx < NC*NST; idx += 32) {
      const int r = idx >> 5, cc = idx & 31;
      kws[idx] = Rs[r*34 + cc];
    }
    if (lane < NC) kws[NC*NST + lane] = Rs[lane*34 + 32];

    // ---- V' = Qxx + Qxu K   (Kᵀ(Qux + Quu K) == 0 identically) ----
    for (int ti = 0; ti < 2; ++ti)
      for (int tj = 0; tj < 2; ++tj) {
        v8f acc = load_tile(Qs, NS, 16*ti, 16*tj, lm, lh);      // Qxx tile
#pragma unroll
        for (int kc = 0; kc < 4; ++kc) {
          v2f a  = ldA(Qs, NS, 16*ti, NST + 4*kc, lm, lh);      // Qxu chunk
          v2f bb = ldB(Rs, 34, 4*kc, 16*tj, lm, lh);            // K chunk
          acc = wmma4(a, bb, acc);
        }
        store_tile(Vt, NST, 16*ti, 16*tj, lm, lh, acc);
      }
    __syncthreads();

    // ---- symmetrize V; v' = qx + Qxu k ----
    for (int idx = lane; idx < NST*NST; idx += 32) {
      const int r = idx >> 5, cc = idx & 31;
      Vs[idx] = 0.5f*(Vt[r*NST + cc] + Vt[cc*NST + r]);
    }
    if (lane < NST) {
      float s = qv[lane];
      for (int j = 0; j < NC; ++j) s += Qs[lane*NS + NST + j] * Rs[j*34 + 32];
      vv[lane] = s;
    }
    __syncthreads();
  }

  // ================= forward rollout =================
  if (lane < NST) xv[lane] = x0g[(size_t)b*NST + lane];
  __syncthreads();

  for (int t = 0; t < Tn; ++t) {
    const float* kws = ws + ((size_t)b*Tn + t)*KSTRIDE;
    if (lane < NC) {                       // u = K x + k
      float s = kws[NC*NST + lane];
      for (int i = 0; i < NST; ++i) s += kws[lane*NST + i] * xv[i];
      xu[NST + lane] = s;
    }
    if (lane < NST) xu[lane] = xv[lane];
    __syncthreads();

    float* tp = tau + ((size_t)t*Bn + b)*NS;
    tp[lane] = xu[lane];
    if (lane < NC) tp[NST + lane] = xu[NST + lane];

    if (t < Tn - 1) {                      // x' = F [x;u] + f
      const float* Ft = Fg + ((size_t)t*Bn + b)*NST*NS;
      const float* ft = fg + ((size_t)t*Bn + b)*NST;
      if (lane < NST) {
        float s = ft[lane];
        for (int j = 0; j < NS; ++j) s += Ft[lane*NS + j] * xu[j];
        xv[lane] = s;
      }
      __syncthreads();
    }
  }
}

extern "C" void kernel_launch(void* const* d_in, const int* in_sizes, int n_in,
                              void* d_out, int out_size, void* d_ws, size_t ws_size,
                              hipStream_t stream) {
  (void)in_sizes; (void)n_in; (void)out_size; (void)ws_size;
  const float* C  = (const float*)d_in[0];
  const float* c  = (const float*)d_in[1];
  const float* F  = (const float*)d_in[2];
  const float* f  = (const float*)d_in[3];
  const float* x0 = (const float*)d_in[4];
  float* tau = (float*)d_out;
  float* ws  = (float*)d_ws;   // needs Bn*Tn*KSTRIDE*4 = ~27 MB
  lqr_riccati_kernel<<<Bn, 32, 0, stream>>>(C, c, F, f, x0, tau, ws);
}